// DualThresholdSelfregulatingIntegrate_45097156608495
// MI455X (gfx1250) — compile-verified
//
#include <hip/hip_runtime.h>

// ---------------------------------------------------------------------------
// Integrate-and-fire scan, [B=32, S=512, D=1024].
//   rates = relu(x)*dt;  per step: 4x { v += r; sp = floor(v); v -= sp }
//   out = sp / dt
// Memory-bound (134 MB @ 23.3 TB/s ~ 6us). Data movement via CDNA5 TDM:
// tensor_load_to_lds DMAs 32-step x 64-float tiles into 6-deep ring of LDS
// buffers (48 KB), tracked with TENSORcnt: 5 tiles in flight per wave
// => ~20 MB of prefetch across 512 blocks, covering HBM latency*BW.
// Single-wave workgroups => no barriers needed; the wave owns its TENSORcnt.
// ---------------------------------------------------------------------------

typedef unsigned int       u32;
typedef unsigned long long u64;
typedef u32 v4u __attribute__((ext_vector_type(4)));
typedef int v4i __attribute__((ext_vector_type(4)));
typedef int v8i __attribute__((ext_vector_type(8)));

#define B_TOT   32
#define S_TOT   512
#define D_TOT   1024
#define CHUNK_D 64                       // d-columns per block
#define CHUNKS  (D_TOT / CHUNK_D)        // 16
#define STEPS   32                       // time steps per DMA tile
#define NI      (S_TOT / STEPS)          // 16 tiles per block
#define NBUFS   6                        // ring depth (5 tiles in flight ahead)
#define BUF_ELEMS (STEPS * CHUNK_D)      // 2048 floats
#define BUF_BYTES (BUF_ELEMS * 4)        // 8 KB
#define LDS_BYTES (NBUFS * BUF_BYTES)    // 48 KB

// Issue one TDM DMA: 2D tile (STEPS rows x CHUNK_D floats) from global -> LDS.
// D# per CDNA5 ISA ch.8 (group0: flags/lds/global addr; group1: dims/strides).
// clang-23 (amdgpu-toolchain) builtin arity: 6 args
//   (uint32x4 g0, int32x8 g1, int32x4 g2, int32x4 g3, int32x8 gx, i32 cpol)
__device__ __forceinline__ void tdm_issue(const float* gtile, u32 lds_off) {
  u64 ga = (u64)gtile;
  v4u g0 = { 1u,                                  // count=1 (valid), user mode
             lds_off,                             // LDS byte address (dyn-LDS base 0)
             (u32)(ga & 0xFFFFFFFFull),           // global_addr[31:0]
             (u32)((ga >> 32) & 0x1FFFFFFull)     // global_addr[56:32]
               | (2u << 30) };                    // type = 2 ("image")
  v8i g1 = { (int)0x00020000u,                    // data_size = 2 (4 bytes)
             (int)((u32)D_TOT << 16),             // tensor_dim0 = 1024 (lo16 @ dw1[31:16])
             (int)((u32)S_TOT << 16),             // tensor_dim1 = 512  (lo16 @ dw2[31:16])
             (int)((u32)CHUNK_D << 16),           // tile_dim0 = 64     (@ dw3[31:16])
             (int)STEPS,                          // tile_dim1 = 32; tile_dim2 = 0
             (int)D_TOT,                          // tensor_dim0_stride = 1024 elems
             0, 0 };                              // dim1_stride unused (2D tile)
  v4i g2 = { 1, 1, 0, 0 };                        // benign dims for unused ranks
  v4i g3 = { 0, 1 << 16, 0, 0 };
  v8i gx = { 0, 0, 0, 0, 0, 0, 0, 0 };            // zero-filled extra group
  __builtin_amdgcn_tensor_load_to_lds(g0, g1, g2, g3, gx, /*cpol=*/0);
}

__device__ __forceinline__ void process_tile(const float* __restrict__ smem,
                                             int buf, float* __restrict__ outRow,
                                             int lane, float& va, float& vb) {
  const float dt     = 0.001f;
  const float inv_dt = 1.0f / 0.001f;             // constant-folded reciprocal
  const float* bufp = smem + buf * BUF_ELEMS;
#pragma unroll 4
  for (int t = 0; t < STEPS; ++t) {
    float2 x = ((const float2*)(bufp + t * CHUNK_D))[lane];  // ds_load_b64
    float ra = fmaxf(x.x, 0.0f) * dt;
    float rb = fmaxf(x.y, 0.0f) * dt;
    float spa, spb;
#pragma unroll
    for (int k = 0; k < 4; ++k) {                 // two interleaved chains (ILP=2)
      va += ra; spa = floorf(va); va -= spa;
      vb += rb; spb = floorf(vb); vb -= spb;
    }
    float2 o; o.x = spa * inv_dt; o.y = spb * inv_dt;
    *(float2*)(outRow + (size_t)t * D_TOT) = o;   // global_store_b64, coalesced
  }
}

extern "C" __global__ void __launch_bounds__(32)
snn_tdm_scan_kernel(const float* __restrict__ in,
                    const float* __restrict__ v0,
                    float* __restrict__ out) {
  const int blk   = blockIdx.x;
  const int b     = blk >> 4;                     // / CHUNKS
  const int chunk = blk & (CHUNKS - 1);
  const int lane  = threadIdx.x;                  // wave32, single-wave block
  const int d0    = chunk * CHUNK_D + lane * 2;

  extern __shared__ float smem[];                 // 48 KB dynamic LDS, base offset 0

  const float* inBase  = in  + (size_t)b * S_TOT * D_TOT + chunk * CHUNK_D;
  float*       outBase = out + (size_t)b * S_TOT * D_TOT + d0;

  // Prologue: fill the 6-deep ring (6 DMAs in flight).
#pragma unroll
  for (int i = 0; i < NBUFS; ++i)
    tdm_issue(inBase + (size_t)i * STEPS * D_TOT, (u32)(i * BUF_BYTES));

  float2 v = *(const float2*)(v0 + (size_t)b * D_TOT + d0);
  float va = v.x, vb = v.y;

  // Steady state: DMA i complete when TENSORcnt <= NBUFS-1 (in-order per wave).
#pragma unroll 1
  for (int i = 0; i < NI - NBUFS; ++i) {          // i = 0..9
    __builtin_amdgcn_s_wait_tensorcnt(NBUFS - 1);
    process_tile(smem, i % NBUFS, outBase + (size_t)i * STEPS * D_TOT, lane, va, vb);
    asm volatile("s_wait_dscnt 0" ::: "memory");  // WAR guard before buffer reuse
    tdm_issue(inBase + (size_t)(i + NBUFS) * STEPS * D_TOT,
              (u32)((i % NBUFS) * BUF_BYTES));
  }
  // Drain tail: i = NI-6 .. NI-1, all 16 DMAs already issued (in-order counts).
#define TAIL(IDX, CNT)                                                         \
  __builtin_amdgcn_s_wait_tensorcnt(CNT);                                      \
  process_tile(smem, (IDX) % NBUFS, outBase + (size_t)(IDX) * STEPS * D_TOT,   \
               lane, va, vb);
  TAIL(NI - 6, 5)
  TAIL(NI - 5, 4)
  TAIL(NI - 4, 3)
  TAIL(NI - 3, 2)
  TAIL(NI - 2, 1)
  TAIL(NI - 1, 0)
#undef TAIL
}

extern "C" void kernel_launch(void* const* d_in, const int* in_sizes, int n_in,
                              void* d_out, int out_size, void* d_ws, size_t ws_size,
                              hipStream_t stream) {
  const float* in  = (const float*)d_in[0];   // [32, 512, 1024] f32
  const float* v0  = (const float*)d_in[1];   // [32, 1024] f32
  float*       out = (float*)d_out;           // [32, 512, 1024] f32

  dim3 grid(B_TOT * CHUNKS);                  // 512 single-wave blocks
  dim3 block(32);
  hipLaunchKernelGGL(snn_tdm_scan_kernel, grid, block, LDS_BYTES, stream,
                     in, v0, out);
}